// Pool3d_54640573939791
// MI455X (gfx1250) — compile-verified
//
#include <hip/hip_runtime.h>
#include <float.h>
#include <math.h>

// Ragged segment max-pool for MI455X (gfx1250, wave32).
// out[p] = max over inputs[nn_index[j]] for j in segment p.  C = 128 fp32.
//
// Roofline: gather stream = TOTAL*512B ~ 819 MB from the 102.4 MB input table
// (L2-resident in the 192 MB L2) -> L2-BW bound.  HBM floor ~160 MB (~7 us).
// Design: 1 wave per output point (lane = 4 channels, one coalesced
// global_load_b128 = 512 B/row), block = 8 points, block's contiguous
// nn_index slice staged into LDS via CDNA5 async-to-LDS (ASYNCcnt), output
// written non-temporally so the gather table stays hot in L2.

typedef float v4f __attribute__((ext_vector_type(4)));
typedef int   v4i __attribute__((ext_vector_type(4)));

#define POINTS_PER_BLOCK 8
#define SIDX_CAP 1024  // staged index capacity (ints) per block; 4 KB LDS

// ---------------- scan kernels: exclusive prefix sum of nn_count -----------

__global__ __launch_bounds__(256) void scan_phase1(const int* __restrict__ cnt,
                                                   int* __restrict__ off,
                                                   int* __restrict__ blockSums,
                                                   int mp) {
  __shared__ int sh[256];
  const int t = threadIdx.x;
  const int base = blockIdx.x * 1024 + t * 4;
  int c0 = 0, c1 = 0, c2 = 0, c3 = 0;
  const bool fullBlock = (blockIdx.x * 1024 + 1024 <= (unsigned)mp);
  if (fullBlock) {
    // counts are streamed exactly once: non-temporal 128-bit load
    v4i c = __builtin_nontemporal_load((const v4i*)(cnt + base));
    c0 = c.x; c1 = c.y; c2 = c.z; c3 = c.w;
  } else {
    if (base + 0 < mp) c0 = cnt[base + 0];
    if (base + 1 < mp) c1 = cnt[base + 1];
    if (base + 2 < mp) c2 = cnt[base + 2];
    if (base + 3 < mp) c3 = cnt[base + 3];
  }
  const int s = c0 + c1 + c2 + c3;
  sh[t] = s;
  __syncthreads();
  int acc = s;
  for (int d = 1; d < 256; d <<= 1) {
    int v = (t >= d) ? sh[t - d] : 0;
    __syncthreads();
    acc += v;
    sh[t] = acc;
    __syncthreads();
  }
  const int excl = acc - s;  // exclusive prefix of thread sums within block
  if (fullBlock) {
    v4i o = {excl, excl + c0, excl + c0 + c1, excl + c0 + c1 + c2};
    *(v4i*)(off + base) = o;
  } else {
    if (base + 0 < mp) off[base + 0] = excl;
    if (base + 1 < mp) off[base + 1] = excl + c0;
    if (base + 2 < mp) off[base + 2] = excl + c0 + c1;
    if (base + 3 < mp) off[base + 3] = excl + c0 + c1 + c2;
  }
  if (t == 255) blockSums[blockIdx.x] = acc;  // block total
}

__global__ __launch_bounds__(1024) void scan_phase2(const int* __restrict__ blockSums,
                                                    int* __restrict__ blockOffsets,
                                                    int nb) {
  __shared__ int sh[1024];
  const int t = threadIdx.x;
  const int s = (t < nb) ? blockSums[t] : 0;
  sh[t] = s;
  __syncthreads();
  int acc = s;
  for (int d = 1; d < 1024; d <<= 1) {
    int v = (t >= d) ? sh[t - d] : 0;
    __syncthreads();
    acc += v;
    sh[t] = acc;
    __syncthreads();
  }
  if (t < nb) blockOffsets[t] = acc - s;  // exclusive scan of block totals
}

__global__ __launch_bounds__(256) void scan_phase3(int* __restrict__ off,
                                                   const int* __restrict__ blockOffsets,
                                                   int mp) {
  const int boff = blockOffsets[blockIdx.x];
  const int base = blockIdx.x * 1024 + threadIdx.x * 4;
  if (blockIdx.x * 1024 + 1024 <= (unsigned)mp) {
    v4i o = *(const v4i*)(off + base);
    o.x += boff; o.y += boff; o.z += boff; o.w += boff;
    *(v4i*)(off + base) = o;
  } else {
#pragma unroll
    for (int k = 0; k < 4; ++k)
      if (base + k < mp) off[base + k] += boff;
  }
}

// ---------------- main gather + segment-max kernel -------------------------

// Inner reduction over one segment; instantiated once with an LDS-derived
// pointer (infer-address-space -> ds_load) and once with a global pointer.
__device__ __forceinline__ v4f pool_rows(const int* __restrict__ ip, int n,
                                         const v4f* __restrict__ in, int lane) {
  v4f a0 = {-INFINITY, -INFINITY, -INFINITY, -INFINITY};
  v4f a1 = a0;
  int j = 0;
  for (; j + 4 <= n; j += 4) {
    const int i0 = ip[j], i1 = ip[j + 1], i2 = ip[j + 2], i3 = ip[j + 3];
    v4f v0 = in[(size_t)i0 * 32 + lane];
    v4f v1 = in[(size_t)i1 * 32 + lane];
    v4f v2 = in[(size_t)i2 * 32 + lane];
    v4f v3 = in[(size_t)i3 * 32 + lane];
    a0 = __builtin_elementwise_max(a0, v0);
    a1 = __builtin_elementwise_max(a1, v1);
    a0 = __builtin_elementwise_max(a0, v2);
    a1 = __builtin_elementwise_max(a1, v3);
  }
  for (; j < n; ++j) {
    const int i0 = ip[j];
    a0 = __builtin_elementwise_max(a0, in[(size_t)i0 * 32 + lane]);
  }
  return __builtin_elementwise_max(a0, a1);
}

__global__ __launch_bounds__(256) void pool_max_kernel(
    const v4f* __restrict__ in,        // [N_IN][32] of float4 (C=128)
    const int* __restrict__ cnt,       // [MP]
    const int* __restrict__ nn_index,  // [TOTAL]
    const int* __restrict__ off,       // [MP] exclusive starts
    v4f* __restrict__ out,             // [MP][32]
    int mp, int total) {
  __shared__ int sIdx[SIDX_CAP];

  const int p0 = blockIdx.x * POINTS_PER_BLOCK;
  const int pend = (p0 + POINTS_PER_BLOCK < mp) ? p0 + POINTS_PER_BLOCK : mp;
  const int blockStart = off[p0];
  const int blockEnd = (pend < mp) ? off[pend] : total;
  const int ndw = blockEnd - blockStart;
  const bool useLds = (ndw <= SIDX_CAP);

  if (useLds) {
    // CDNA5 async copy: stage this block's contiguous index slice into LDS.
    const unsigned ldsBase = (unsigned)(size_t)(&sIdx[0]);
    for (int t = threadIdx.x; t < ndw; t += 256) {
      const unsigned laddr = ldsBase + 4u * (unsigned)t;
      const int* g = nn_index + blockStart + t;
      asm volatile("global_load_async_to_lds_b32 %0, %1, off"
                   :: "v"(laddr), "v"(g)
                   : "memory");
    }
    asm volatile("s_wait_asynccnt 0x0" ::: "memory");
  }
  __syncthreads();

  const int wave = threadIdx.x >> 5;  // wave32
  const int lane = threadIdx.x & 31;
  const int p = p0 + wave;
  if (p >= mp) return;

  const int start = off[p];
  const int n = cnt[p];

  v4f r;
  if (useLds) {
    r = pool_rows(sIdx + (start - blockStart), n, in, lane);  // ds_load path
  } else {
    r = pool_rows(nn_index + start, n, in, lane);             // global fallback
  }

  // Output has zero reuse: non-temporal store keeps the gather table hot in L2.
  __builtin_nontemporal_store(r, &out[(size_t)p * 32 + lane]);
}

// ---------------- host launch ----------------------------------------------

extern "C" void kernel_launch(void* const* d_in, const int* in_sizes, int n_in,
                              void* d_out, int out_size, void* d_ws, size_t ws_size,
                              hipStream_t stream) {
  const float* inputs = (const float*)d_in[0];    // [N_IN*128] fp32
  const int* nn_count = (const int*)d_in[1];      // [MP]
  const int* nn_index = (const int*)d_in[2];      // [TOTAL]
  const int mp = in_sizes[1];
  const int total = in_sizes[2];

  const int nb = (mp + 1023) / 1024;  // scan blocks (<=1024 supported)
  int* off = (int*)d_ws;              // [mp] exclusive segment starts
  int* blockSums = off + mp;          // [nb]
  int* blockOffsets = blockSums + nb; // [nb]

  scan_phase1<<<nb, 256, 0, stream>>>(nn_count, off, blockSums, mp);
  scan_phase2<<<1, 1024, 0, stream>>>(blockSums, blockOffsets, nb);
  scan_phase3<<<nb, 256, 0, stream>>>(off, blockOffsets, mp);

  const int nblocks = (mp + POINTS_PER_BLOCK - 1) / POINTS_PER_BLOCK;
  pool_max_kernel<<<nblocks, 256, 0, stream>>>(
      (const v4f*)inputs, nn_count, nn_index, off, (v4f*)d_out, mp, total);
}